// SMPLBodyModel_25383256719613
// MI455X (gfx1250) — compile-verified
//
#include <hip/hip_runtime.h>
#include <math.h>

// ---------------------------------------------------------------------------
// SMPL forward on MI455X (gfx1250).
//   outputs (concatenated in d_out): joints (B,24,3) | vertices (B,6890,3) |
//                                    glob_rotmats (B,24,3,3)
// ---------------------------------------------------------------------------

#define NJ   24
#define NBET 10
#define NV   6890
#define NCH  (NV * 3)          // 20670 vertex-channels
#define KDIM 217               // 10 betas + 207 pose features
#define KPAD 224               // zero-padded K (multiple of 8)
#define LPITCH 228             // LDS pitch for feature tile (bank-conflict free)
#define NPAD 20736             // 81 * 256: guard-free N for the packed B panel

typedef __attribute__((ext_vector_type(2))) float v2f;
typedef __attribute__((ext_vector_type(8))) float v8f;

__device__ constexpr int kParents[NJ] = {
    -1, 0, 0, 0, 1, 2, 3, 4, 5, 6, 7, 8, 9, 9, 9, 12, 13, 14, 16, 17, 18, 19, 20, 21};

// ---------------------------------------------------------------------------
// Kernel A: per-body Rodrigues, kinematic chain, joint outputs, feature pack,
// and per-joint affine G = [R | pos - R*jloc + trans].  One wave per body.
// ---------------------------------------------------------------------------
__global__ void __launch_bounds__(32)
smpl_prep_kernel(const float* __restrict__ pose,        // (B,24,3)
                 const float* __restrict__ betas,       // (B,10)
                 const float* __restrict__ trans,       // (B,3)
                 const float* __restrict__ J_template,  // (24,3)
                 const float* __restrict__ J_shapedirs, // (24,3,10)
                 float* __restrict__ out_joints,        // (B,24,3)
                 float* __restrict__ out_glob,          // (B,24,9)
                 float* __restrict__ feat,              // (B,KPAD) workspace
                 float* __restrict__ G)                 // (B,24,12) workspace
{
    const int b    = blockIdx.x;
    const int lane = threadIdx.x;   // 0..31

    __shared__ float rel [NJ][9];
    __shared__ float glob[NJ][9];
    __shared__ float jloc[NJ][3];
    __shared__ float pos [NJ][3];
    __shared__ float betas_s[NBET];

    if (lane < NBET) betas_s[lane] = betas[b * NBET + lane];
    __syncthreads();

    if (lane < NJ) {
        const float rx = pose[(b * NJ + lane) * 3 + 0];
        const float ry = pose[(b * NJ + lane) * 3 + 1];
        const float rz = pose[(b * NJ + lane) * 3 + 2];
        const float theta = sqrtf(rx * rx + ry * ry + rz * rz);
        const float st = (theta < 1e-8f) ? 1.0f : theta;
        const float ax = rx / st, ay = ry / st, az = rz / st;
        const float s = sinf(theta);
        const float oc = 1.0f - cosf(theta);
        rel[lane][0] = 1.0f + oc * (-(ay * ay + az * az));
        rel[lane][1] = -s * az + oc * ax * ay;
        rel[lane][2] =  s * ay + oc * ax * az;
        rel[lane][3] =  s * az + oc * ax * ay;
        rel[lane][4] = 1.0f + oc * (-(ax * ax + az * az));
        rel[lane][5] = -s * ax + oc * ay * az;
        rel[lane][6] = -s * ay + oc * ax * az;
        rel[lane][7] =  s * ax + oc * ay * az;
        rel[lane][8] = 1.0f + oc * (-(ax * ax + ay * ay));
        #pragma unroll
        for (int c = 0; c < 3; ++c) {
            float v = J_template[lane * 3 + c];
            #pragma unroll
            for (int sIdx = 0; sIdx < NBET; ++sIdx)
                v += J_shapedirs[(lane * 3 + c) * NBET + sIdx] * betas_s[sIdx];
            jloc[lane][c] = v;
        }
    }
    __syncthreads();

    if (lane == 0) {
        #pragma unroll
        for (int e = 0; e < 9; ++e) glob[0][e] = rel[0][e];
        #pragma unroll
        for (int c = 0; c < 3; ++c) pos[0][c] = jloc[0][c];
        #pragma unroll
        for (int i = 1; i < NJ; ++i) {
            const int p = kParents[i];
            #pragma unroll
            for (int r = 0; r < 3; ++r)
                #pragma unroll
                for (int c = 0; c < 3; ++c)
                    glob[i][r * 3 + c] = glob[p][r * 3 + 0] * rel[i][0 * 3 + c] +
                                         glob[p][r * 3 + 1] * rel[i][1 * 3 + c] +
                                         glob[p][r * 3 + 2] * rel[i][2 * 3 + c];
            const float bx = jloc[i][0] - jloc[p][0];
            const float by = jloc[i][1] - jloc[p][1];
            const float bz = jloc[i][2] - jloc[p][2];
            #pragma unroll
            for (int r = 0; r < 3; ++r)
                pos[i][r] = pos[p][r] + glob[p][r * 3 + 0] * bx +
                                        glob[p][r * 3 + 1] * by +
                                        glob[p][r * 3 + 2] * bz;
        }
    }
    __syncthreads();

    const float tvec[3] = {trans[b * 3 + 0], trans[b * 3 + 1], trans[b * 3 + 2]};

    for (int i = lane; i < NJ * 3; i += 32)
        out_joints[b * NJ * 3 + i] = pos[i / 3][i % 3] + tvec[i % 3];

    for (int i = lane; i < NJ * 9; i += 32)
        out_glob[b * NJ * 9 + i] = glob[i / 9][i % 9];

    for (int k = lane; k < KPAD; k += 32) {
        float f;
        if (k < NBET)      f = betas_s[k];
        else if (k < KDIM) { const int q = k - NBET; f = rel[1 + q / 9][q % 9]; }
        else               f = 0.0f;
        feat[b * KPAD + k] = f;
    }

    if (lane < NJ) {
        const int j = lane;
        float* g = G + ((size_t)b * NJ + j) * 12;
        #pragma unroll
        for (int e = 0; e < 9; ++e) g[e] = glob[j][e];
        #pragma unroll
        for (int r = 0; r < 3; ++r)
            g[9 + r] = pos[j][r] + tvec[r]
                       - (glob[j][r * 3 + 0] * jloc[j][0] +
                          glob[j][r * 3 + 1] * jloc[j][1] +
                          glob[j][r * 3 + 2] * jloc[j][2]);
    }
}

// ---------------------------------------------------------------------------
// Kernel P: one-time pack of [shapedirs | posedirs] into K-pair-major,
// zero-padded Dirs2[KPAD/2][NPAD] of float2: Dirs2[k>>1][n] = (d[k,n], d[k+1,n]).
// A lane's B-fragment (two adjacent K values) then becomes one b64 load.
// ---------------------------------------------------------------------------
__global__ void __launch_bounds__(256)
smpl_pack_dirs(const float* __restrict__ shapedirs,  // (NCH, 10)
               const float* __restrict__ posedirs,   // (NCH, 207)
               float2* __restrict__ Dirs2)           // (KPAD/2, NPAD) float2
{
    __shared__ float tile[64][LPITCH];   // 58.4 KB (one-shot kernel)
    const int n0  = blockIdx.x * 64;
    const int tid = threadIdx.x;

    for (int i = tid; i < 64 * KPAD; i += 256) {
        const int r = i / KPAD, k = i % KPAD;
        const int n = n0 + r;
        float v = 0.0f;
        if (n < NCH) {
            if (k < NBET)       v = shapedirs[(size_t)n * NBET + k];
            else if (k < KDIM)  v = posedirs[(size_t)n * 207 + (k - NBET)];
        }
        tile[r][k] = v;
    }
    __syncthreads();

    for (int i = tid; i < (KPAD / 2) * 64; i += 256) {
        const int kp = i / 64, c = i % 64;
        Dirs2[(size_t)kp * NPAD + n0 + c] =
            make_float2(tile[c][2 * kp], tile[c][2 * kp + 1]);
    }
}

// ---------------------------------------------------------------------------
// Kernel B (fast path): blendshape GEMM via V_WMMA_F32_16X16X4_F32 on the
// packed K-pair panel.  Branch-free inner loop per K-step:
//   1x ds_load_b64 (A frag) + 2x coalesced global_load_b64 (B, two N-tiles)
//   + 2x WMMA.
//   M = B (tiles of 16), N tile per block = 256 (8 waves x 2 x 16), K = 224.
// Fragment layouts (wave32, ISA 7.12.2):
//   A 16x4 f32 : lane L -> M = L%16, VGPR{0,1} = K = k0+2*(L/16)+{0,1}
//   B 4x16 f32 : lane L -> N = L%16, VGPR{0,1} = K = k0+2*(L/16)+{0,1}
//   C/D 16x16  : VGPR r -> M = r + 8*(L/16), N = L%16
// ---------------------------------------------------------------------------
__global__ void __launch_bounds__(256)
smpl_blendshape_wmma(const float* __restrict__ feat,       // (B, KPAD)
                     const float2* __restrict__ Dirs2,     // (KPAD/2, NPAD)
                     const float* __restrict__ v_template, // (NCH)
                     float* __restrict__ vout)             // (B, NCH) = v_posed
{
    __shared__ float feat_s[16 * LPITCH];   // 14.25 KB

    const int tid  = threadIdx.x;
    const int b0   = blockIdx.x * 16;
    const int wave = tid >> 5;
    const int lane = tid & 31;
    const int nl   = lane & 15;
    const int half = lane >> 4;
    const int n0   = blockIdx.y * 256 + wave * 32 + nl;  // first N-tile column
    const int n1   = n0 + 16;                            // second N-tile column

    for (int i = tid; i < 16 * KPAD; i += 256) {
        const int row = i / KPAD, col = i % KPAD;
        feat_s[row * LPITCH + col] = feat[(size_t)(b0 + row) * KPAD + col];
    }
    __syncthreads();

    v8f acc0 = {0.f, 0.f, 0.f, 0.f, 0.f, 0.f, 0.f, 0.f};
    v8f acc1 = {0.f, 0.f, 0.f, 0.f, 0.f, 0.f, 0.f, 0.f};
    const float*  arow = feat_s + nl * LPITCH + 2 * half;
    const float2* bp   = Dirs2 + (size_t)half * NPAD;    // lane's K-pair offset

    #pragma unroll 8
    for (int k0 = 0; k0 < KPAD; k0 += 4) {
        v2f a;
        a.x = arow[k0];
        a.y = arow[k0 + 1];

        const float2* bk = bp + (size_t)(k0 >> 1) * NPAD;
        const float2  q0 = bk[n0];
        const float2  q1 = bk[n1];
        v2f bf0, bf1;
        bf0.x = q0.x;  bf0.y = q0.y;
        bf1.x = q1.x;  bf1.y = q1.y;

        acc0 = __builtin_amdgcn_wmma_f32_16x16x4_f32(
            false, a, false, bf0, (short)0, acc0, false, false);
        acc1 = __builtin_amdgcn_wmma_f32_16x16x4_f32(
            false, a, false, bf1, (short)0, acc1, false, false);
    }

    // + v_template: N is fixed per lane, one scalar feeds all 8 rows.
    const bool v0ok = (n0 < NCH);
    const bool v1ok = (n1 < NCH);
    const float vt0 = v0ok ? v_template[n0] : 0.0f;
    const float vt1 = v1ok ? v_template[n1] : 0.0f;
    #pragma unroll
    for (int r = 0; r < 8; ++r) { acc0[r] += vt0; acc1[r] += vt1; }

    const int mbase = b0 + half * 8;
    if (v0ok) {
        #pragma unroll
        for (int r = 0; r < 8; ++r)
            vout[(size_t)(mbase + r) * NCH + n0] = acc0[r];
    }
    if (v1ok) {
        #pragma unroll
        for (int r = 0; r < 8; ++r)
            vout[(size_t)(mbase + r) * NCH + n1] = acc1[r];
    }
}

// ---------------------------------------------------------------------------
// Kernel B (fallback, used only if ws_size cannot hold the packed panel):
// direct gather from shapedirs/posedirs (divergent but correct).
// ---------------------------------------------------------------------------
__global__ void __launch_bounds__(256)
smpl_blendshape_wmma_direct(const float* __restrict__ feat,
                            const float* __restrict__ shapedirs,
                            const float* __restrict__ posedirs,
                            const float* __restrict__ v_template,
                            float* __restrict__ vout)
{
    __shared__ float feat_s[16 * LPITCH];

    const int tid  = threadIdx.x;
    const int b0   = blockIdx.x * 16;
    const int wave = tid >> 5;
    const int lane = tid & 31;
    const int nl   = lane & 15;
    const int half = lane >> 4;
    const int n    = blockIdx.y * 128 + wave * 16 + nl;

    for (int i = tid; i < 16 * KPAD; i += 256) {
        const int row = i / KPAD, col = i % KPAD;
        feat_s[row * LPITCH + col] = feat[(size_t)(b0 + row) * KPAD + col];
    }
    __syncthreads();

    const bool  nvalid = (n < NCH);
    const float* srow  = shapedirs + (size_t)(nvalid ? n : 0) * NBET;
    const float* prow  = posedirs  + (size_t)(nvalid ? n : 0) * 207;

    v8f acc = {0.f, 0.f, 0.f, 0.f, 0.f, 0.f, 0.f, 0.f};
    const float* arow = feat_s + nl * LPITCH;

    for (int k0 = 0; k0 < KDIM; k0 += 4) {
        const int kk = k0 + 2 * half;
        v2f a;
        a.x = arow[kk];
        a.y = arow[kk + 1];
        v2f bf;
        float b0v = (kk     < NBET) ? srow[kk]
                  : (kk     < KDIM) ? prow[kk - NBET]     : 0.0f;
        float b1v = (kk + 1 < NBET) ? srow[kk + 1]
                  : (kk + 1 < KDIM) ? prow[kk + 1 - NBET] : 0.0f;
        bf.x = nvalid ? b0v : 0.0f;
        bf.y = nvalid ? b1v : 0.0f;
        acc = __builtin_amdgcn_wmma_f32_16x16x4_f32(
            false, a, false, bf, (short)0, acc, false, false);
    }

    const float vt = nvalid ? v_template[n] : 0.0f;
    #pragma unroll
    for (int r = 0; r < 8; ++r) acc[r] += vt;

    if (nvalid) {
        const int mbase = b0 + half * 8;
        #pragma unroll
        for (int r = 0; r < 8; ++r)
            vout[(size_t)(mbase + r) * NCH + n] = acc[r];
    }
}

// ---------------------------------------------------------------------------
// Kernel C: linear blend skinning, in place on the vertices region.
// ---------------------------------------------------------------------------
__global__ void __launch_bounds__(256)
smpl_lbs_kernel(const float* __restrict__ G,       // (B,24,12)
                const float* __restrict__ weights, // (NV,24), 16B-aligned rows
                float* __restrict__ verts)         // (B,NV,3): v_posed in, vertices out
{
    __shared__ float Gs[NJ * 12];
    const int b   = blockIdx.y;
    const int tid = threadIdx.x;

    for (int i = tid; i < NJ * 12; i += 256) Gs[i] = G[(size_t)b * NJ * 12 + i];
    __syncthreads();

    const int v = blockIdx.x * 256 + tid;
    if (v >= NV) return;

    float R[9] = {0, 0, 0, 0, 0, 0, 0, 0, 0};
    float T[3] = {0, 0, 0};
    const float4* wp = (const float4*)(weights + (size_t)v * NJ);  // 96B rows
    #pragma unroll
    for (int q = 0; q < 6; ++q) {
        const float4 w4 = wp[q];
        const float wl[4] = {w4.x, w4.y, w4.z, w4.w};
        #pragma unroll
        for (int u = 0; u < 4; ++u) {
            const int j = q * 4 + u;
            const float w = wl[u];
            #pragma unroll
            for (int e = 0; e < 9; ++e) R[e] += w * Gs[j * 12 + e];
            T[0] += w * Gs[j * 12 + 9];
            T[1] += w * Gs[j * 12 + 10];
            T[2] += w * Gs[j * 12 + 11];
        }
    }

    float* vp = verts + (size_t)b * NCH + (size_t)v * 3;
    const float px = vp[0], py = vp[1], pz = vp[2];
    vp[0] = R[0] * px + R[1] * py + R[2] * pz + T[0];
    vp[1] = R[3] * px + R[4] * py + R[5] * pz + T[1];
    vp[2] = R[6] * px + R[7] * py + R[8] * pz + T[2];
}

// ---------------------------------------------------------------------------
extern "C" void kernel_launch(void* const* d_in, const int* in_sizes, int n_in,
                              void* d_out, int out_size, void* d_ws, size_t ws_size,
                              hipStream_t stream)
{
    const float* pose        = (const float*)d_in[0];
    const float* betas       = (const float*)d_in[1];
    const float* trans       = (const float*)d_in[2];
    const float* v_template  = (const float*)d_in[3];
    const float* shapedirs   = (const float*)d_in[4];
    const float* posedirs    = (const float*)d_in[5];
    const float* J_template  = (const float*)d_in[6];
    const float* J_shapedirs = (const float*)d_in[7];
    const float* weights     = (const float*)d_in[8];

    const int B = in_sizes[0] / (NJ * 3);   // 1024

    float* out        = (float*)d_out;
    float* out_joints = out;                               // B*72
    float* out_verts  = out + (size_t)B * NJ * 3;          // B*20670
    float* out_glob   = out_verts + (size_t)B * NCH;       // B*216

    // workspace layout: feat (B*KPAD) | G (B*288) | Dirs2 (KPAD/2 * NPAD float2)
    float*  feat  = (float*)d_ws;
    float*  G     = feat + (size_t)B * KPAD;
    float2* Dirs2 = (float2*)(G + (size_t)B * NJ * 12);

    const size_t smallBytes = ((size_t)B * KPAD + (size_t)B * NJ * 12) * sizeof(float);
    const size_t needBytes  = smallBytes + (size_t)KPAD * NPAD * sizeof(float);

    smpl_prep_kernel<<<dim3(B), dim3(32), 0, stream>>>(
        pose, betas, trans, J_template, J_shapedirs,
        out_joints, out_glob, feat, G);

    if (ws_size >= needBytes) {
        smpl_pack_dirs<<<dim3(NPAD / 64), dim3(256), 0, stream>>>(
            shapedirs, posedirs, Dirs2);
        dim3 gridB(B / 16, NPAD / 256);      // 64 x 81
        smpl_blendshape_wmma<<<gridB, dim3(256), 0, stream>>>(
            feat, Dirs2, v_template, out_verts);
    } else {
        dim3 gridB(B / 16, (NCH + 127) / 128);   // 64 x 162
        smpl_blendshape_wmma_direct<<<gridB, dim3(256), 0, stream>>>(
            feat, shapedirs, posedirs, v_template, out_verts);
    }

    dim3 gridC((NV + 255) / 256, B);         // 27 x 1024
    smpl_lbs_kernel<<<gridC, dim3(256), 0, stream>>>(G, weights, out_verts);
}